// WorldNormalShader_2860448219468
// MI455X (gfx1250) — compile-verified
//
#include <hip/hip_runtime.h>
#include <stdint.h>

// ---------------------------------------------------------------------------
// pytorch3d-style world-normal shading + softmax_rgb_blend for MI455X/gfx1250.
//
// Roofline: ~218 MB of streaming traffic per call -> ~9.4 us floor at
// 23.3 TB/s HBM; ~0.9 GFLOP of VALU -> pure bandwidth problem. The
// per-fragment interpolation is a batched (1x3)x(3x3) matvec with a
// *different* 3x3 per fragment, so WMMA has no shared operand tile and is
// inapplicable. CDNA5-specific paths used instead:
//   * GLOBAL_LOAD_ASYNC_TO_LDS_B128 staging of the streaming fragment
//     tensors (ASYNCcnt + s_wait_asynccnt), read back with conflict-free
//     b128 DS loads.
//   * L2-resident face->vertex-normal table (4.8 MB, reused 8.4M times)
//     collapsing the p2f->faces->verts double indirection into 3 b128
//     gathers; output written with non-temporal stores so the streaming
//     results don't evict the table from the 192 MB L2.
// ---------------------------------------------------------------------------

#define KF   8      // fragments per pixel
#define PPB  256    // pixels per block -> 2048-fragment tile per block

#define SIGMA_INV   1.0e4f          // 1/1e-4
#define GAMMA_INV   1.0e4f          // 1/1e-4
#define ZFAR_C      100.0f
#define ZRANGE_INV  (1.0f / 99.0f)  // 1/(zfar - znear)
#define EPS_P       1e-10f

typedef int   v4i __attribute__((vector_size(16)));
typedef float v4f __attribute__((ext_vector_type(4)));
typedef __attribute__((address_space(1))) v4i* gptr_v4i;
typedef __attribute__((address_space(3))) v4i* lptr_v4i;

#if __has_builtin(__builtin_amdgcn_global_load_async_to_lds_b128)
#define ASYNC_VIA_BUILTIN 1
#else
#define ASYNC_VIA_BUILTIN 0
#warning "gfx1250 path: falling back to inline-asm global_load_async_to_lds_b128"
#endif

__device__ __forceinline__ void async_copy16(const void* g, void* l) {
#if ASYNC_VIA_BUILTIN
  __builtin_amdgcn_global_load_async_to_lds_b128(
      (gptr_v4i)(void*)g, (lptr_v4i)l, /*imm offset=*/0, /*cpol=*/0);
#else
  uint32_t lds_off =
      (uint32_t)(size_t)(__attribute__((address_space(3))) void*)l;
  uint64_t gaddr = (uint64_t)(size_t)g;
  asm volatile("global_load_async_to_lds_b128 %0, %1, off"
               :: "v"(lds_off), "v"(gaddr)
               : "memory");
#endif
}

__device__ __forceinline__ void wait_async_zero() {
#if __has_builtin(__builtin_amdgcn_s_wait_asynccnt)
  __builtin_amdgcn_s_wait_asynccnt(0);
#else
  asm volatile("s_wait_asynccnt 0" ::: "memory");
#endif
}

// ---------------- shared compute body ---------------------------------------

template <bool USE_TABLE>
__device__ __forceinline__ void shade_blend_body(
    const float* __restrict__ bary, const float* __restrict__ zbuf,
    const float* __restrict__ dists, const float* __restrict__ verts,
    const int* __restrict__ p2f, const int* __restrict__ faces,
    const float* __restrict__ fvn, float* __restrict__ out, int nPix) {
  __shared__ float s_bary[PPB * KF * 3];  // 24576 B
  __shared__ float s_z[PPB * KF];         //  8192 B
  __shared__ float s_d[PPB * KF];         //  8192 B
  __shared__ int   s_f[PPB * KF];         //  8192 B

  const int t = threadIdx.x;
  const int pixBase = blockIdx.x * PPB;
  const long fragBase = (long)pixBase * KF;

  // ---- async DMA stage of this block's fragment tile into LDS ----
  {
    const float4* gb = (const float4*)(bary + fragBase * 3);
    const float4* gz = (const float4*)(zbuf + fragBase);
    const float4* gd = (const float4*)(dists + fragBase);
    const int4*   gf = (const int4*)(p2f + fragBase);
    float4* lb = (float4*)s_bary;
    float4* lz = (float4*)s_z;
    float4* ld = (float4*)s_d;
    int4*   lf = (int4*)s_f;
#pragma unroll
    for (int i = 0; i < 6; ++i)  // 1536 float4 of bary across 256 lanes
      async_copy16(gb + t + PPB * i, lb + t + PPB * i);
#pragma unroll
    for (int i = 0; i < 2; ++i) {  // 512 x 16B each for z / d / p2f
      async_copy16(gz + t + PPB * i, lz + t + PPB * i);
      async_copy16(gd + t + PPB * i, ld + t + PPB * i);
      async_copy16(gf + t + PPB * i, lf + t + PPB * i);
    }
  }
  wait_async_zero();   // my lanes' DMA done
  __syncthreads();     // everyone's DMA done

  const int pix = pixBase + t;
  if (pix >= nPix) return;

  // ---- pull this pixel's tile slice into registers with b128 DS loads ----
  // (per-lane spans are 16B aligned: bary 96B @ 96*t, z/d/f 32B @ 32*t)
  union { float4 v[6]; float f[24]; } B;
  union { float4 v[2]; float f[8];  } Z, D;
  union { int4   v[2]; int   f[8];  } Fc;
#pragma unroll
  for (int i = 0; i < 6; ++i) B.v[i] = ((const float4*)s_bary)[t * 6 + i];
#pragma unroll
  for (int i = 0; i < 2; ++i) {
    Z.v[i]  = ((const float4*)s_z)[t * 2 + i];
    D.v[i]  = ((const float4*)s_d)[t * 2 + i];
    Fc.v[i] = ((const int4*)s_f)[t * 2 + i];
  }

  float prob[KF], zinv[KF];
  float cr[KF], cg[KF], cb[KF];
  float zmax = EPS_P;

#pragma unroll
  for (int k = 0; k < KF; ++k) {
    const int f = Fc.f[k];
    const bool live = (f >= 0);
    const int fi = live ? f : 0;

    float n0x, n0y, n0z, n1x, n1y, n1z, n2x, n2y, n2z;
    if (USE_TABLE) {
      const float4* r = (const float4*)(fvn + (size_t)fi * 12);
      float4 a = r[0], b = r[1], c = r[2];
      n0x = a.x; n0y = a.y; n0z = a.z;
      n1x = b.x; n1y = b.y; n1z = b.z;
      n2x = c.x; n2y = c.y; n2z = c.z;
    } else {
      const int* fp = faces + 3 * (size_t)fi;
      const int v0 = fp[0], v1 = fp[1], v2 = fp[2];
      const float* q0 = verts + 3 * (size_t)v0;
      const float* q1 = verts + 3 * (size_t)v1;
      const float* q2 = verts + 3 * (size_t)v2;
      n0x = q0[0]; n0y = q0[1]; n0z = q0[2];
      n1x = q1[0]; n1y = q1[1]; n1z = q1[2];
      n2x = q2[0]; n2y = q2[1]; n2z = q2[2];
    }

    const float b0 = B.f[k * 3 + 0];
    const float b1 = B.f[k * 3 + 1];
    const float b2 = B.f[k * 3 + 2];
    const float m = live ? 1.0f : 0.0f;
    cr[k] = m * (b0 * n0x + b1 * n1x + b2 * n2x);
    cg[k] = m * (b0 * n0y + b1 * n1y + b2 * n2y);
    cb[k] = m * (b0 * n0z + b1 * n1z + b2 * n2z);

    // prob = sigmoid(-d/sigma) * mask = mask / (1 + exp(d/sigma))
    prob[k] = live ? (1.0f / (1.0f + __expf(D.f[k] * SIGMA_INV))) : 0.0f;

    const float zi = live ? (ZFAR_C - Z.f[k]) * ZRANGE_INV : 0.0f;
    zinv[k] = zi;
    zmax = fmaxf(zmax, zi);
  }

  const float delta = __expf((EPS_P - zmax) * GAMMA_INV);
  float denom = delta;
  float ar = 0.f, ag = 0.f, ab = 0.f, om = 1.0f;
#pragma unroll
  for (int k = 0; k < KF; ++k) {
    const float w = prob[k] * __expf((zinv[k] - zmax) * GAMMA_INV);
    denom += w;
    ar += w * cr[k];
    ag += w * cg[k];
    ab += w * cb[k];
    om *= (1.0f - prob[k]);
  }
  const float inv = 1.0f / denom;
  const float wbg = delta * inv;  // background = (1,1,1)

  v4f o;
  o.x = ar * inv + wbg;
  o.y = ag * inv + wbg;
  o.z = ab * inv + wbg;
  o.w = 1.0f - om;
  // Non-temporal store: keep the 16.8 MB of output from evicting the
  // L2-resident fvn gather table.
  __builtin_nontemporal_store(o, ((v4f*)out) + pix);
}

// ---------------- kernels (table variant first for disasm visibility) -------

__global__ __launch_bounds__(PPB) void shade_blend_table_kernel(
    const float* __restrict__ bary, const float* __restrict__ zbuf,
    const float* __restrict__ dists, const float* __restrict__ verts,
    const int* __restrict__ p2f, const int* __restrict__ faces,
    const float* __restrict__ fvn, float* __restrict__ out, int nPix) {
  shade_blend_body<true>(bary, zbuf, dists, verts, p2f, faces, fvn, out, nPix);
}

__global__ __launch_bounds__(PPB) void shade_blend_gather_kernel(
    const float* __restrict__ bary, const float* __restrict__ zbuf,
    const float* __restrict__ dists, const float* __restrict__ verts,
    const int* __restrict__ p2f, const int* __restrict__ faces,
    float* __restrict__ out, int nPix) {
  shade_blend_body<false>(bary, zbuf, dists, verts, p2f, faces, nullptr, out,
                          nPix);
}

// fvn[f] = { n(v0).xyz,_ , n(v1).xyz,_ , n(v2).xyz,_ }  (12 floats, 48 B)
__global__ __launch_bounds__(256) void build_fvn_kernel(
    const float* __restrict__ verts,   // [V,3]
    const int*   __restrict__ faces,   // [F,3]
    float*       __restrict__ fvn,     // [F,12]
    int F) {
  int f = blockIdx.x * blockDim.x + threadIdx.x;
  if (f >= F) return;
  int v0 = faces[3 * f + 0];
  int v1 = faces[3 * f + 1];
  int v2 = faces[3 * f + 2];
  float4* o = (float4*)(fvn + (size_t)f * 12);
  o[0] = make_float4(verts[3 * v0], verts[3 * v0 + 1], verts[3 * v0 + 2], 0.f);
  o[1] = make_float4(verts[3 * v1], verts[3 * v1 + 1], verts[3 * v1 + 2], 0.f);
  o[2] = make_float4(verts[3 * v2], verts[3 * v2 + 1], verts[3 * v2 + 2], 0.f);
}

// ---------------- launch ----------------------------------------------------

extern "C" void kernel_launch(void* const* d_in, const int* in_sizes, int n_in,
                              void* d_out, int out_size, void* d_ws,
                              size_t ws_size, hipStream_t stream) {
  const float* bary  = (const float*)d_in[0];
  const float* zbuf  = (const float*)d_in[1];
  const float* dists = (const float*)d_in[2];
  const float* verts = (const float*)d_in[3];
  const int*   p2f   = (const int*)d_in[4];
  const int*   faces = (const int*)d_in[5];
  float*       out   = (float*)d_out;

  const int nPix = in_sizes[1] / KF;       // zbuf is [P,K]
  const int F    = in_sizes[5] / 3;        // faces is [F,3]
  const int grid = (nPix + PPB - 1) / PPB;

  const size_t tblBytes = (size_t)F * 12 * sizeof(float);
  if (ws_size >= tblBytes) {
    float* fvn = (float*)d_ws;
    build_fvn_kernel<<<(F + 255) / 256, 256, 0, stream>>>(verts, faces, fvn, F);
    shade_blend_table_kernel<<<grid, PPB, 0, stream>>>(
        bary, zbuf, dists, verts, p2f, faces, fvn, out, nPix);
  } else {
    shade_blend_gather_kernel<<<grid, PPB, 0, stream>>>(
        bary, zbuf, dists, verts, p2f, faces, out, nPix);
  }
}